// G2A_76433238000124
// MI455X (gfx1250) — compile-verified
//
#include <hip/hip_runtime.h>
#include <math.h>

typedef __attribute__((ext_vector_type(2))) float v2f;
typedef __attribute__((ext_vector_type(8))) float v8f;

#define Nn 3
#define Hh 2048
#define Aa 512
#define Ff 42
#define TAUc 0.01f

__constant__ int IDXc[3][2] = {{1,2},{0,2},{0,1}};

// ---------------------------------------------------------------------------
// Skinny-GEMV via WMMA f32 16x16x4, K known at compile time.
//   Out(Mvalid, Nout) = A(Mvalid, K) @ W(Nout, K)^T + bias
// One workgroup = 16 output columns; 8 waves split K; software-pipelined
// 8-WMMA chunks (32 K-values) with double-buffered b64 loads; LDS reduce.
// ---------------------------------------------------------------------------
template <int K>
__global__ __launch_bounds__(256) void gemv16_wmma(
    const float* __restrict__ Ain,   // (Mvalid, K) row-major
    const float* __restrict__ W,     // (Nout, K) row-major
    const float* __restrict__ bias,  // (Nout) or nullptr
    float* __restrict__ Out,         // (Mvalid, Nout)
    int Nout, int Mvalid)
{
    constexpr int KCH  = K / 8;      // per-wave K range
    constexpr int ITER = KCH / 32;   // 8 WMMAs per iteration
    static_assert(ITER >= 2 && (ITER % 2) == 0, "pipeline needs even ITER");

    const int n0    = blockIdx.x << 4;
    const int wave  = threadIdx.x >> 5;   // 0..7
    const int lane  = threadIdx.x & 31;
    const int lhalf = lane >> 4;          // 0 / 1 -> K offset 0 / 2
    const int l16   = lane & 15;

    // A operand: lane holds row m = lane%16 (clamp padded rows -> row 0, unused)
    const int arow = (l16 < Mvalid) ? l16 : 0;
    const float* aPtr = Ain + (size_t)arow * K + wave * KCH + (lhalf << 1);
    // B operand: lane holds weight row (output neuron) n0 + lane%16
    const float* wPtr = W + (size_t)(n0 + l16) * K + wave * KCH + (lhalf << 1);

    v2f a[2][8], b[2][8];

    // prologue: stage 0
    #pragma unroll
    for (int u = 0; u < 8; ++u) {
        a[0][u] = *(const v2f*)(aPtr + 4 * u);
        b[0][u] = *(const v2f*)(wPtr + 4 * u);
    }

    v8f c0 = {}, c1 = {};
    #pragma unroll 2
    for (int it = 0; it < ITER; ++it) {
        const int cur = it & 1, nxt = cur ^ 1;
        if (it + 1 < ITER) {
            const float* ap = aPtr + 32;
            const float* wp = wPtr + 32;
            #pragma unroll
            for (int u = 0; u < 8; ++u) {
                a[nxt][u] = *(const v2f*)(ap + 4 * u);
                b[nxt][u] = *(const v2f*)(wp + 4 * u);
            }
        }
        #pragma unroll
        for (int u = 0; u < 8; u += 2) {
            c0 = __builtin_amdgcn_wmma_f32_16x16x4_f32(
                     false, a[cur][u],     false, b[cur][u],     (short)0, c0, false, false);
            c1 = __builtin_amdgcn_wmma_f32_16x16x4_f32(
                     false, a[cur][u + 1], false, b[cur][u + 1], (short)0, c1, false, false);
        }
        aPtr += 32;
        wPtr += 32;
    }
    v8f c = c0 + c1;

    // C layout: VGPR v, lanes 0-15 -> M=v  (rows >=8 live in lanes 16-31, unused)
    __shared__ float red[8][6][16];
    if (lhalf == 0) {
        #pragma unroll
        for (int v = 0; v < 6; ++v) red[wave][v][l16] = c[v];
    }
    __syncthreads();

    const int t = threadIdx.x;
    if (t < Mvalid * 16) {
        const int mm = t >> 4;
        const int nn = t & 15;
        float s = 0.f;
        #pragma unroll
        for (int wv = 0; wv < 8; ++wv) s += red[wv][mm][nn];
        if (bias) s += bias[n0 + nn];
        Out[(size_t)mm * Nout + n0 + nn] = s;
    }
}

// ---------------------------------------------------------------------------
// Encoder: x = relu(obs @ W_enc^T + b_enc)   (3 x 2048, K = 42 -> scalar dot)
// ---------------------------------------------------------------------------
__global__ __launch_bounds__(256) void enc_kernel(
    const float* __restrict__ obs, const float* __restrict__ W,
    const float* __restrict__ b, float* __restrict__ x)
{
    int t = blockIdx.x * blockDim.x + threadIdx.x;
    if (t >= Nn * Hh) return;
    int m = t / Hh, n = t % Hh;
    float s = b[n];
    const float* o = obs + m * Ff;
    const float* w = W + (size_t)n * Ff;
    #pragma unroll
    for (int k = 0; k < Ff; ++k) s = fmaf(o[k], w[k], s);
    x[t] = fmaxf(s, 0.f);
}

// ---------------------------------------------------------------------------
// GRU gate combine: h = (1-z)*n + z*hprev
//   gi: (M, 3H) ; gh: (M, 3H) or bias vector (3H) when gh_is_bias
//   hprev nullptr => zeros (first scan step)
// ---------------------------------------------------------------------------
__global__ __launch_bounds__(256) void gru_combine(
    const float* __restrict__ gi, const float* __restrict__ gh, int gh_is_bias,
    const float* __restrict__ hprev, float* __restrict__ hnew,
    float* __restrict__ hnew2, int M)
{
    int t = blockIdx.x * blockDim.x + threadIdx.x;
    if (t >= M * Hh) return;
    int m = t / Hh, j = t % Hh;
    size_t gib = (size_t)m * (3 * Hh);
    size_t ghb = gh_is_bias ? 0 : (size_t)m * (3 * Hh);
    float ir = gi[gib + j], iz = gi[gib + Hh + j], in = gi[gib + 2 * Hh + j];
    float hr = gh[ghb + j], hz = gh[ghb + Hh + j], hn = gh[ghb + 2 * Hh + j];
    float r  = 1.f / (1.f + __expf(-(ir + hr)));
    float z  = 1.f / (1.f + __expf(-(iz + hz)));
    float nn = tanhf(in + r * hn);
    float hp = hprev ? hprev[t] : 0.f;
    float h  = (1.f - z) * nn + z * hp;
    hnew[t] = h;
    if (hnew2) hnew2[t] = h;
}

// ---------------------------------------------------------------------------
// Build seq (6, 2H): row r = t*3 + i  ->  concat(h[i], h[IDX[i][t]])
// ---------------------------------------------------------------------------
__global__ __launch_bounds__(256) void seq_build(
    const float* __restrict__ h, float* __restrict__ seq)
{
    int t = blockIdx.x * blockDim.x + threadIdx.x;
    if (t >= 6 * 2 * Hh) return;
    int row = t / (2 * Hh), col = t % (2 * Hh);
    int tt = row / 3, i = row % 3;
    int src = (col < Hh) ? i : IDXc[i][tt];
    int c2  = (col < Hh) ? col : col - Hh;
    seq[t] = h[src * Hh + c2];
}

// ---------------------------------------------------------------------------
// Finalization (single block): logits -> gumbel softmax -> attention -> xx,
// plus the obs-distance flag.  outs_f/outs_r are (T=2, N=3, H).
// ---------------------------------------------------------------------------
__device__ __forceinline__ float block_reduce256(float v, float* sred)
{
    sred[threadIdx.x] = v;
    __syncthreads();
    for (int s = 128; s > 0; s >>= 1) {
        if ((int)threadIdx.x < s) sred[threadIdx.x] += sred[threadIdx.x + s];
        __syncthreads();
    }
    float r = sred[0];
    __syncthreads();
    return r;
}

__global__ __launch_bounds__(256) void final_kernel(
    const float* __restrict__ outs_f, const float* __restrict__ outs_r,
    const float* __restrict__ W_hard, const float* __restrict__ b_hard,
    const float* __restrict__ gumbel, const float* __restrict__ q,
    const float* __restrict__ k, const float* __restrict__ obs,
    float* __restrict__ out)
{
    __shared__ float sred[256];
    __shared__ float logits[6][2];
    __shared__ float hw[3][2];
    __shared__ float sc[3][2];

    // logits: rows ordered (i, t);  h_hard[i*2+t] = [outs_f[t][i], outs_r[1-t][i]]
    for (int row = 0; row < 6; ++row) {
        int i = row >> 1, tt = row & 1;
        const float* hf = outs_f + ((size_t)tt * 3 + i) * Hh;
        const float* hr = outs_r + ((size_t)(1 - tt) * 3 + i) * Hh;
        for (int c = 0; c < 2; ++c) {
            const float* w = W_hard + (size_t)c * (2 * Hh);
            float p = 0.f;
            for (int j = threadIdx.x; j < Hh; j += 256) p = fmaf(hf[j], w[j], p);
            for (int j = threadIdx.x; j < Hh; j += 256) p = fmaf(hr[j], w[Hh + j], p);
            p = block_reduce256(p, sred);
            if (threadIdx.x == 0) logits[row][c] = p + b_hard[c];
        }
    }
    __syncthreads();

    if (threadIdx.x < 6) {
        int row = threadIdx.x, i = row >> 1, tt = row & 1;
        float l0 = (logits[row][0] + gumbel[row * 2 + 0]) / TAUc;
        float l1 = (logits[row][1] + gumbel[row * 2 + 1]) / TAUc;
        float mx = fmaxf(l0, l1);
        float e0 = __expf(l0 - mx), e1 = __expf(l1 - mx);
        hw[i][tt] = e1 / (e0 + e1);
    }
    __syncthreads();

    // attention scores: q[i] . k[IDX[i][j]] / sqrt(A)
    for (int i = 0; i < 3; ++i)
        for (int j = 0; j < 2; ++j) {
            const float* qq = q + (size_t)i * Aa;
            const float* kk = k + (size_t)IDXc[i][j] * Aa;
            float p = 0.f;
            for (int a = threadIdx.x; a < Aa; a += 256) p = fmaf(qq[a], kk[a], p);
            p = block_reduce256(p, sred);
            if (threadIdx.x == 0) sc[i][j] = p * (1.f / sqrtf((float)Aa));
        }
    __syncthreads();

    if (threadIdx.x < 3) {
        int i = threadIdx.x;
        float s0 = sc[i][0], s1 = sc[i][1];
        float mx = fmaxf(s0, s1);
        float e0 = __expf(s0 - mx), e1 = __expf(s1 - mx);
        float d  = e0 + e1;
        out[Nn * Hh + i * 2 + 0] = (e0 / d) * hw[i][0];
        out[Nn * Hh + i * 2 + 1] = (e1 / d) * hw[i][1];
    }

    if (threadIdx.x == 0) {
        float res[3][2]; int jadj[3];
        for (int i = 0; i < 3; ++i) {
            for (int jj = 0; jj < 2; ++jj) {
                int o = IDXc[i][jj];
                float s = 0.f;
                for (int kx = 0; kx < Ff; ++kx) {
                    float d2 = obs[i * Ff + kx] - obs[o * Ff + kx];
                    s = fmaf(d2, d2, s);
                }
                res[i][jj] = s;
            }
            int jl = (res[i][1] > res[i][0]) ? 1 : 0;  // argmax, first-max wins
            float mv = res[i][jl];
            int ja = jl;
            if (mv != 0.f && ja >= i) ja += 1;
            jadj[i] = ja;
        }
        int tmp[3][2];
        for (int i = 0; i < 3; ++i) {
            tmp[i][0] = (i < jadj[i]) ? i : jadj[i];
            tmp[i][1] = (i > jadj[i]) ? i : jadj[i];
        }
        bool eq01 = tmp[0][0] == tmp[1][0] && tmp[0][1] == tmp[1][1];
        bool eq02 = tmp[0][0] == tmp[2][0] && tmp[0][1] == tmp[2][1];
        bool eq12 = tmp[1][0] == tmp[2][0] && tmp[1][1] == tmp[2][1];
        int f0 = -1, f1 = -1;
        if (eq12)              { f0 = tmp[1][0]; f1 = tmp[1][1]; }
        else if (eq01 || eq02) { f0 = tmp[0][0]; f1 = tmp[0][1]; }
        out[Nn * Hh + 6 + 0] = (float)f0;
        out[Nn * Hh + 6 + 1] = (float)f1;
    }
}

// ---------------------------------------------------------------------------
extern "C" void kernel_launch(void* const* d_in, const int* in_sizes, int n_in,
                              void* d_out, int out_size, void* d_ws, size_t ws_size,
                              hipStream_t stream)
{
    const float* obs    = (const float*)d_in[0];
    const float* hidden = (const float*)d_in[1];
    const float* gumbel = (const float*)d_in[2];
    const float* W_enc  = (const float*)d_in[3];
    const float* b_enc  = (const float*)d_in[4];
    const float* Wih_c  = (const float*)d_in[5];
    const float* Whh_c  = (const float*)d_in[6];
    const float* bih_c  = (const float*)d_in[7];
    const float* bhh_c  = (const float*)d_in[8];
    const float* Wih_f  = (const float*)d_in[9];
    const float* Whh_f  = (const float*)d_in[10];
    const float* bih_f  = (const float*)d_in[11];
    const float* bhh_f  = (const float*)d_in[12];
    const float* Wih_r  = (const float*)d_in[13];
    const float* Whh_r  = (const float*)d_in[14];
    const float* bih_r  = (const float*)d_in[15];
    const float* bhh_r  = (const float*)d_in[16];
    const float* W_hard = (const float*)d_in[17];
    const float* b_hard = (const float*)d_in[18];
    const float* Wq     = (const float*)d_in[19];
    const float* Wk     = (const float*)d_in[20];
    float* out = (float*)d_out;

    // workspace layout (floats)
    float* ws = (float*)d_ws;
    float* x      = ws;                 // 3*2048
    float* h_out  = x      + 6144;      // 3*2048
    float* gi_c   = h_out  + 6144;      // 3*6144
    float* gh_c   = gi_c   + 18432;     // 3*6144
    float* seq    = gh_c   + 18432;     // 6*4096
    float* gi_f   = seq    + 24576;     // 6*6144
    float* gi_r   = gi_f   + 36864;     // 6*6144
    float* outs_f = gi_r   + 36864;     // 2*3*2048
    float* outs_r = outs_f + 12288;     // 2*3*2048
    float* gh_f1  = outs_r + 12288;     // 3*6144
    float* gh_r1  = gh_f1  + 18432;     // 3*6144
    float* qbuf   = gh_r1  + 18432;     // 3*512
    float* kbuf   = qbuf   + 1536;      // 3*512

    const int G3H = 3 * Hh;             // 6144 gate width

    // 1) encoder
    enc_kernel<<<24, 256, 0, stream>>>(obs, W_enc, b_enc, x);

    // 2) ctrl GRU gemms (one HBM pass each over Wih_c / Whh_c)
    gemv16_wmma<Hh><<<G3H / 16, 256, 0, stream>>>(x,      Wih_c, bih_c, gi_c, G3H, 3);
    gemv16_wmma<Hh><<<G3H / 16, 256, 0, stream>>>(hidden, Whh_c, bhh_c, gh_c, G3H, 3);

    // 3) ctrl GRU combine -> h_out (also first 6144 floats of d_out)
    gru_combine<<<24, 256, 0, stream>>>(gi_c, gh_c, 0, hidden, h_out, out, 3);

    // 4) seq gather (6, 2H)
    seq_build<<<96, 256, 0, stream>>>(h_out, seq);

    // 5) both scan timesteps batched: one pass each over Wih_f / Wih_r (M=6, K=2H)
    gemv16_wmma<2 * Hh><<<G3H / 16, 256, 0, stream>>>(seq, Wih_f, bih_f, gi_f, G3H, 6);
    gemv16_wmma<2 * Hh><<<G3H / 16, 256, 0, stream>>>(seq, Wih_r, bih_r, gi_r, G3H, 6);

    // 6) q / k projections
    gemv16_wmma<Hh><<<Aa / 16, 256, 0, stream>>>(h_out, Wq, nullptr, qbuf, Aa, 3);
    gemv16_wmma<Hh><<<Aa / 16, 256, 0, stream>>>(h_out, Wk, nullptr, kbuf, Aa, 3);

    // 7) step 0 (carry h0 = 0 => gh is just the bhh bias; no Whh GEMM)
    //    forward uses seq[t=0] (gi_f rows 0..2); reverse uses seq[t=1] (gi_r rows 3..5)
    gru_combine<<<24, 256, 0, stream>>>(gi_f,           bhh_f, 1, nullptr, outs_f, nullptr, 3);
    gru_combine<<<24, 256, 0, stream>>>(gi_r + 3 * G3H, bhh_r, 1, nullptr, outs_r, nullptr, 3);

    // 8) step 1 hidden-path gemms (one pass each over Whh_f / Whh_r)
    gemv16_wmma<Hh><<<G3H / 16, 256, 0, stream>>>(outs_f, Whh_f, bhh_f, gh_f1, G3H, 3);
    gemv16_wmma<Hh><<<G3H / 16, 256, 0, stream>>>(outs_r, Whh_r, bhh_r, gh_r1, G3H, 3);

    // 9) step 1 combines
    gru_combine<<<24, 256, 0, stream>>>(gi_f + 3 * G3H, gh_f1, 0, outs_f, outs_f + 3 * Hh, nullptr, 3);
    gru_combine<<<24, 256, 0, stream>>>(gi_r,           gh_r1, 0, outs_r, outs_r + 3 * Hh, nullptr, 3);

    // 10) logits / gumbel softmax / attention / flag
    final_kernel<<<1, 256, 0, stream>>>(outs_f, outs_r, W_hard, b_hard, gumbel,
                                        qbuf, kbuf, obs, out);
}